// FeatureTransformer_42795054137554
// MI455X (gfx1250) — compile-verified
//
#include <hip/hip_runtime.h>

// Sparse feature transformer (NNUE-style) for MI455X / gfx1250.
// out[b,:] = sum_f value[b,f] * W[idx[b,f],:]  (idx>=0 only)  + bias
//
// Design (compile-only reasoning from MI455X specs):
//  * Weight table = 84 MB -> resident in the 192 MB L2; gathers are L2 hits,
//    so the problem is gather *latency*, not bandwidth.
//  * CDNA5 async global->LDS DMA (ASYNCcnt) gathers each 2 KB row with zero
//    VGPR pressure; 32 gathers kept in flight per wave.
//  * Thread t gathers bytes [16t,16t+16) of every row == exactly the bytes it
//    later consumes, so no cross-lane LDS dependency exists: no barrier between
//    issue and consume, and per-wave staged waits (asynccnt 24/16/8/0) overlap
//    FMA work on early rows with in-flight gathers of later rows.
//  * Invalid (-1) slots: clamp index to 0, force coeff to 0.0 -> uniform
//    fixed-trip loops, no serial compaction chain (round-2 asm showed the
//    scalarized thread-0 compaction was a ~64-deep dependent s_load chain).

#define OUT_D     512
#define MAX_FEATS 32
#define THREADS   128   // 4 waves of 32; OUT_D/4 == THREADS

typedef float f4 __attribute__((ext_vector_type(4)));
typedef int   i4 __attribute__((ext_vector_type(4)));
typedef __attribute__((address_space(1))) i4 gi4;   // global int4
typedef __attribute__((address_space(3))) i4 li4;   // LDS int4

#if __has_builtin(__builtin_amdgcn_global_load_async_to_lds_b128) && \
    __has_builtin(__builtin_amdgcn_s_wait_asynccnt)
#define USE_ASYNC_LDS 1
#else
#define USE_ASYNC_LDS 0
#warning "gfx1250 async-to-LDS builtins unavailable; using synchronous fallback"
#endif

__global__ __launch_bounds__(THREADS)
void FeatureTransformer_gather_kernel(const int* __restrict__ fidx,
                                      const float* __restrict__ fval,
                                      const float* __restrict__ weight,
                                      const float* __restrict__ bias,
                                      float* __restrict__ out)
{
    __shared__ float s_rows[MAX_FEATS][OUT_D];   // 64 KB of gathered rows
    __shared__ int   s_cidx[MAX_FEATS];          // clamped indices
    __shared__ float s_coef[MAX_FEATS];          // coeff (0.0 for invalid)

    const int b = blockIdx.x;
    const int t = threadIdx.x;

    // --- parallel staging: lanes 0..31 each handle one feature slot ---
    if (t < MAX_FEATS) {
        const int   ix = fidx[(size_t)b * MAX_FEATS + t];
        const float v  = fval[(size_t)b * MAX_FEATS + t];
        const bool ok  = (ix >= 0);
        s_cidx[t] = ok ? ix : 0;
        s_coef[t] = ok ? v : 0.0f;
    }
    __syncthreads();   // staging visible to all 4 waves

    // --- issue all 32 async row-gathers (thread t moves 16 B per row) ---
#if USE_ASYNC_LDS
#pragma unroll
    for (int f = 0; f < MAX_FEATS; ++f) {
        const float* g = weight + (size_t)s_cidx[f] * OUT_D + t * 4;
        __builtin_amdgcn_global_load_async_to_lds_b128(
            (gi4*)g, (li4*)&s_rows[f][t * 4], /*offset=*/0, /*cpol=*/0);
    }
#else
#pragma unroll
    for (int f = 0; f < MAX_FEATS; ++f) {
        const f4 r = *(const f4*)(weight + (size_t)s_cidx[f] * OUT_D + t * 4);
        *(f4*)&s_rows[f][t * 4] = r;
    }
#endif

    f4 acc = *(const f4*)(bias + t * 4);   // overlaps with in-flight gathers

    // --- staged consume: async loads retire in order, each thread reads only
    //     its own gathered bytes -> wait on own wave's ASYNCcnt only.
#if USE_ASYNC_LDS
    __builtin_amdgcn_s_wait_asynccnt(24);
#endif
#pragma unroll
    for (int f = 0; f < 8; ++f)
        acc += s_coef[f] * *(const f4*)&s_rows[f][t * 4];

#if USE_ASYNC_LDS
    __builtin_amdgcn_s_wait_asynccnt(16);
#endif
#pragma unroll
    for (int f = 8; f < 16; ++f)
        acc += s_coef[f] * *(const f4*)&s_rows[f][t * 4];

#if USE_ASYNC_LDS
    __builtin_amdgcn_s_wait_asynccnt(8);
#endif
#pragma unroll
    for (int f = 16; f < 24; ++f)
        acc += s_coef[f] * *(const f4*)&s_rows[f][t * 4];

#if USE_ASYNC_LDS
    __builtin_amdgcn_s_wait_asynccnt(0);
#endif
#pragma unroll
    for (int f = 24; f < 32; ++f)
        acc += s_coef[f] * *(const f4*)&s_rows[f][t * 4];

    // non-temporal store: keep the 16.8 MB output stream from evicting the
    // L2-resident weight table
    __builtin_nontemporal_store(acc, (f4*)(out + (size_t)b * OUT_D + t * 4));
}

extern "C" void kernel_launch(void* const* d_in, const int* in_sizes, int n_in,
                              void* d_out, int out_size, void* d_ws, size_t ws_size,
                              hipStream_t stream)
{
    const int*   fidx   = (const int*)d_in[0];    // [B, 32] int32
    const float* fval   = (const float*)d_in[1];  // [B, 32] f32
    const float* weight = (const float*)d_in[2];  // [40960, 512] f32
    const float* bias   = (const float*)d_in[3];  // [512] f32
    float*       out    = (float*)d_out;          // [B, 512] f32

    const int batch = in_sizes[0] / MAX_FEATS;    // 8192

    FeatureTransformer_gather_kernel<<<batch, THREADS, 0, stream>>>(
        fidx, fval, weight, bias, out);
}